// Multi_Headed_40441412059266
// MI455X (gfx1250) — compile-verified
//
#include <hip/hip_runtime.h>
#include <hip/hip_bf16.h>
#include <stdint.h>

typedef __attribute__((ext_vector_type(2))) float v2f;
typedef __attribute__((ext_vector_type(8))) float v8f;
typedef __attribute__((ext_vector_type(2))) int   v2i;

#define NB 32
#define SS 2048
#define DD 80
#define HH 3
#define DKK 20

static __device__ __forceinline__ v8f wmma4(v2f a, v2f b, v8f c) {
  // V_WMMA_F32_16X16X4_F32: D = A(16x4,f32) * B(4x16,f32) + C(16x16,f32)
  return __builtin_amdgcn_wmma_f32_16x16x4_f32(false, a, false, b, (short)0, c,
                                               false, false);
}

typedef __attribute__((address_space(1))) v2i* as1_v2i;
typedef __attribute__((address_space(3))) v2i* as3_v2i;

// One lane-parallel async copy: 8 bytes/lane, global -> LDS, tracked by ASYNCcnt.
static __device__ __forceinline__ void async_b64(const float* g, float* l) {
#if __has_builtin(__builtin_amdgcn_global_load_async_to_lds_b64)
  // Builtin takes typed AS pointers: (<2 x i32> addrspace(1)*, <2 x i32>
  // addrspace(3)*, i32 offset, i32 cpol). Build them via integer round-trip:
  // global flat VA == AS1 VA; LDS flat address carries the LDS byte offset in
  // bits [31:0] (ISA aperture rule).
  __builtin_amdgcn_global_load_async_to_lds_b64(
      (as1_v2i)(uintptr_t)g, (as3_v2i)(uint32_t)(uintptr_t)l, 0, 0);
#else
  unsigned loff = (unsigned)(uintptr_t)l;       // LDS byte offset (low 32 bits)
  uint64_t ga = (uint64_t)(uintptr_t)g;
  asm volatile("global_load_async_to_lds_b64 %0, %1, off"
               :: "v"(loff), "v"(ga) : "memory");
#endif
}

// ---------------------------------------------------------------------------
// Kernel 1: per-head Q/K/V projections. One wave per (b, qkv, h, S-tile of 16).
// ---------------------------------------------------------------------------
__global__ __launch_bounds__(256) void proj_kernel(
    const float* __restrict__ Qin, const float* __restrict__ Kin,
    const float* __restrict__ Vin, const float* __restrict__ Wq,
    const float* __restrict__ Wk, const float* __restrict__ Wv,
    float* __restrict__ Qb, float* __restrict__ Kb, float* __restrict__ Vb) {
  const int lane = threadIdx.x & 31;
  const int half = lane >> 4;
  const int col  = lane & 15;
  int wid = blockIdx.x * (blockDim.x >> 5) + (threadIdx.x >> 5);
  const int mt = wid & 127; wid >>= 7;          // 128 S-tiles
  const int h = wid % 3;    wid /= 3;
  const int qkv = wid % 3;
  const int b = wid / 3;

  const float* X = (qkv == 0) ? Qin : (qkv == 1) ? Kin : Vin;
  const float* W = ((qkv == 0) ? Wq : (qkv == 1) ? Wk : Wv) + h * DD * DKK;
  float* O = ((qkv == 0) ? Qb : (qkv == 1) ? Kb : Vb) +
             (size_t)(b * HH + h) * SS * DKK;

  const int s0 = mt * 16;
  const float* Xr = X + ((size_t)b * SS + s0) * DD;

  v8f acc0 = {}; v8f acc1 = {};
  for (int st = 0; st < 20; ++st) {             // K = 80 -> 20 steps of 4
    const int k = 4 * st + 2 * half;
    v2f a;  a.x  = Xr[(size_t)col * DD + k];   a.y  = Xr[(size_t)col * DD + k + 1];
    v2f b0; b0.x = W[k * DKK + col];           b0.y = W[(k + 1) * DKK + col];
    const int c2 = (16 + col > DKK - 1) ? (DKK - 1) : (16 + col);
    v2f b1; b1.x = W[k * DKK + c2];            b1.y = W[(k + 1) * DKK + c2];
    acc0 = wmma4(a, b0, acc0);
    acc1 = wmma4(a, b1, acc1);
  }
  for (int vg = 0; vg < 8; ++vg) {              // C layout: row = vg + 8*half
    const int r = s0 + vg + 8 * half;
    O[(size_t)r * DKK + col] = acc0[vg];
    if (col < DKK - 16) O[(size_t)r * DKK + 16 + col] = acc1[vg];
  }
}

// ---------------------------------------------------------------------------
// Kernel 2: flash attention with ASYNCcnt double-buffered K/V staging.
// Block = 8 waves; wave w owns queries [qb*128 + w*16, +16) of one (b,h).
// Each K/V tile (16x20 f32 = 1280 contiguous bytes) is staged to LDS by five
// GLOBAL_LOAD_ASYNC_TO_LDS_B64 ops (32 lanes x 8B), prefetching block kb+1
// while block kb runs its 13 WMMAs + online softmax.
// ---------------------------------------------------------------------------
__global__ __launch_bounds__(256) void attn_kernel(
    const float* __restrict__ Qb, const float* __restrict__ Kb,
    const float* __restrict__ Vb, float* __restrict__ Zb) {
  __shared__ __align__(16) float shkv[8 * 2 * 2 * 320]; // [wave][buf][K|V][320]
  __shared__ float plds[8 * 16 * 18];
  const int lane = threadIdx.x & 31;
  const int half = lane >> 4;
  const int col  = lane & 15;
  const int w    = threadIdx.x >> 5;
  const int qb = blockIdx.x & 15;
  const int h  = (blockIdx.x >> 4) % 3;
  const int b  = blockIdx.x / 48;
  const int q0 = qb * 128 + w * 16;

  const size_t base = (size_t)(b * HH + h) * SS;
  const float* Qp = Qb + (base + q0) * DKK;

  v2f aq[5];                                    // Q fragments, all 20 dims
  for (int st = 0; st < 5; ++st) {
    const int k = 4 * st + 2 * half;
    aq[st].x = Qp[col * DKK + k];
    aq[st].y = Qp[col * DKK + k + 1];
  }

  float rowM[8], rowL[8];
  for (int vg = 0; vg < 8; ++vg) { rowM[vg] = -3.0e38f; rowL[vg] = 0.0f; }
  v8f z0 = {}; v8f z1 = {};
  float* lp = &plds[w * 16 * 18];
  const float scale = 0.223606797749979f;       // 1/sqrt(20)

  auto issue_tile = [&](int kb, int buf) {      // stage K+V block kb -> LDS
    const float* gK = Kb + (base + (size_t)kb * 16) * DKK + lane * 2;
    const float* gV = Vb + (base + (size_t)kb * 16) * DKK + lane * 2;
    float* lK = &shkv[((w * 2 + buf) * 2 + 0) * 320] + lane * 2;
    float* lV = &shkv[((w * 2 + buf) * 2 + 1) * 320] + lane * 2;
#pragma unroll
    for (int i = 0; i < 5; ++i) {
      async_b64(gK + i * 64, lK + i * 64);
      async_b64(gV + i * 64, lV + i * 64);
    }
  };

  const int nkb = (q0 >> 4) + 1;                // causal key-block count
  issue_tile(0, 0);
  for (int kb = 0; kb < nkb; ++kb) {
    asm volatile("s_wait_asynccnt 0x0" ::: "memory");  // tile kb resident
    if (kb + 1 < nkb) issue_tile(kb + 1, (kb + 1) & 1); // overlap next fetch
    const float* lk = &shkv[((w * 2 + (kb & 1)) * 2 + 0) * 320];
    const float* lv = &shkv[((w * 2 + (kb & 1)) * 2 + 1) * 320];

    v8f c = {};                                 // scores = Q * K^T (from LDS)
    for (int st = 0; st < 5; ++st) {
      const int k = 4 * st + 2 * half;
      v2f bk; bk.x = lk[col * DKK + k]; bk.y = lk[col * DKK + k + 1];
      c = wmma4(aq[st], bk, c);
    }

    const int sidx = kb * 16 + col;             // scale + causal mask
    for (int vg = 0; vg < 8; ++vg) {
      float sc = c[vg] * scale;
      if (sidx > q0 + vg + 8 * half) sc = -3.0e38f;
      c[vg] = sc;
    }

    // Online softmax: rows live across 16 lanes of a half-wave.
    for (int vg = 0; vg < 8; ++vg) {
      float m = c[vg];
      m = fmaxf(m, __shfl_xor(m, 8, 16));
      m = fmaxf(m, __shfl_xor(m, 4, 16));
      m = fmaxf(m, __shfl_xor(m, 2, 16));
      m = fmaxf(m, __shfl_xor(m, 1, 16));
      const float nm = fmaxf(rowM[vg], m);
      const float alpha = __expf(rowM[vg] - nm);
      rowM[vg] = nm;
      const float p = __expf(c[vg] - nm);
      c[vg] = p;
      float rs = p;
      rs += __shfl_xor(rs, 8, 16);
      rs += __shfl_xor(rs, 4, 16);
      rs += __shfl_xor(rs, 2, 16);
      rs += __shfl_xor(rs, 1, 16);
      rowL[vg] = rowL[vg] * alpha + rs;
      z0[vg] *= alpha;
      z1[vg] *= alpha;
    }

    // Transpose P (C layout -> A layout) through private LDS tile.
    for (int vg = 0; vg < 8; ++vg)
      lp[(vg + 8 * half) * 18 + col] = c[vg];   // lp[q_local][s_local]
    asm volatile("s_wait_dscnt 0x0" ::: "memory");

    for (int st = 0; st < 4; ++st) {            // Z += P * V (V from LDS)
      const int k = 4 * st + 2 * half;
      v2f pa;  pa.x  = lp[col * 18 + k];        pa.y  = lp[col * 18 + k + 1];
      v2f bv0; bv0.x = lv[k * DKK + col];       bv0.y = lv[(k + 1) * DKK + col];
      const int c2 = (16 + col > DKK - 1) ? (DKK - 1) : (16 + col);
      v2f bv1; bv1.x = lv[k * DKK + c2];        bv1.y = lv[(k + 1) * DKK + c2];
      z0 = wmma4(pa, bv0, z0);
      z1 = wmma4(pa, bv1, z1);
    }
    asm volatile("s_wait_dscnt 0x0" ::: "memory");
  }

  // Finalize and store head-concatenated Z: [B, S, H*DK].
  float* Zp = Zb + ((size_t)b * SS + q0) * (HH * DKK) + h * DKK;
  for (int vg = 0; vg < 8; ++vg) {
    const float inv = 1.0f / rowL[vg];
    const int r = vg + 8 * half;
    Zp[(size_t)r * (HH * DKK) + col] = z0[vg] * inv;
    if (col < DKK - 16) Zp[(size_t)r * (HH * DKK) + 16 + col] = z1[vg] * inv;
  }
}

// ---------------------------------------------------------------------------
// Kernel 3: out = Z[B*S,60] x WO[60,60]. One wave per 16-row tile, 4 N-tiles
// reuse the same A fragments; WO is L2-resident.
// ---------------------------------------------------------------------------
__global__ __launch_bounds__(256) void outproj_kernel(
    const float* __restrict__ Zb, const float* __restrict__ WO,
    float* __restrict__ out) {
  const int lane = threadIdx.x & 31;
  const int half = lane >> 4;
  const int col  = lane & 15;
  const int wid = blockIdx.x * (blockDim.x >> 5) + (threadIdx.x >> 5);
  const int m0 = wid * 16;
  const float* Zr = Zb + (size_t)m0 * 60;

  v8f acc[4] = {{}, {}, {}, {}};
  for (int st = 0; st < 15; ++st) {             // K = 60 -> 15 steps
    const int k = 4 * st + 2 * half;
    v2f a; a.x = Zr[col * 60 + k]; a.y = Zr[col * 60 + k + 1];
    for (int nt = 0; nt < 4; ++nt) {
      int n = 16 * nt + col; if (n > 59) n = 59;
      v2f bb; bb.x = WO[k * 60 + n]; bb.y = WO[(k + 1) * 60 + n];
      acc[nt] = wmma4(a, bb, acc[nt]);
    }
  }
  for (int nt = 0; nt < 4; ++nt) {
    const int n = 16 * nt + col;
    if (n < 60) {
      for (int vg = 0; vg < 8; ++vg)
        out[(size_t)(m0 + vg + 8 * half) * 60 + n] = acc[nt][vg];
    }
  }
}

extern "C" void kernel_launch(void* const* d_in, const int* in_sizes, int n_in,
                              void* d_out, int out_size, void* d_ws, size_t ws_size,
                              hipStream_t stream) {
  const float* key_in = (const float*)d_in[0];
  const float* val_in = (const float*)d_in[1];
  const float* qry_in = (const float*)d_in[2];
  const float* Wq = (const float*)d_in[3];
  const float* Wk = (const float*)d_in[4];
  const float* Wv = (const float*)d_in[5];
  const float* WO = (const float*)d_in[6];
  float* out = (float*)d_out;

  // Workspace carve: Q,K,V [B,H,S,20] + Z [B,S,60]  (~63 MB total fp32)
  float* Qb = (float*)d_ws;
  const size_t per = (size_t)NB * HH * SS * DKK;
  float* Kbf = Qb + per;
  float* Vbf = Kbf + per;
  float* Zb  = Vbf + per;

  // 32*3*3*128 = 36864 wave-tasks, 8 waves/block
  proj_kernel<<<4608, 256, 0, stream>>>(qry_in, key_in, val_in, Wq, Wk, Wv,
                                        Qb, Kbf, Vbf);
  // B*H*(S/128) workgroups, 8 query-tiles each
  attn_kernel<<<NB * HH * (SS / 128), 256, 0, stream>>>(Qb, Kbf, Vbf, Zb);
  // (B*S/16) wave-tiles / 8 waves per block
  outproj_kernel<<<(NB * SS / 16) / 8, 256, 0, stream>>>(Zb, WO, out);
}